// STPGCNs_56418690400820
// MI455X (gfx1250) — compile-verified
//
#include <hip/hip_runtime.h>
#include <math.h>

// ---- problem constants ----
#define BB   8
#define TT   12
#define VV   512
#define CC   128
#define DD   8      // gaussian dim d
#define TSZ  3
#define LL   3
#define PP   12
#define BETA 2
#define KTOT 1536   // TSZ*VV
#define NBT  96     // BB*TT

typedef float v2f __attribute__((ext_vector_type(2)));
typedef float v8f __attribute__((ext_vector_type(8)));

__device__ inline v8f wmma4(v2f a, v2f b, v8f c) {
  // D = A(16x4 f32) * B(4x16 f32) + C(16x16 f32)
  return __builtin_amdgcn_wmma_f32_16x16x4_f32(false, a, false, b, (short)0, c, false, false);
}

__device__ inline v8f zero8() {
  v8f r;
#pragma unroll
  for (int i = 0; i < 8; ++i) r[i] = 0.0f;
  return r;
}

__device__ inline float sigmf(float x) { return 1.0f / (1.0f + expf(-x)); }

__device__ inline float gaussd(const float* e, const float* mu, const float* is) {
  float s = 0.0f;
#pragma unroll
  for (int dd = 0; dd < DD; ++dd) {
    float ev = e ? e[dd] : 0.0f;
    float df = ev - mu[dd];
    float g  = is[dd];
    s = fmaf(-0.5f * df * df, g * g, s);
  }
  return s;
}

// ---------------- input layer: h[b,t,v,c] = x[b,t,v,0]*iw[c] + ib[c] ----------------
__global__ void input_kernel(const float* __restrict__ x, const float* __restrict__ iw,
                             const float* __restrict__ ib, float* __restrict__ h) {
  int idx = blockIdx.x * 256 + threadIdx.x;     // < 6291456
  int m = idx >> 7, c = idx & 127;
  h[idx] = fmaf(x[m], iw[c], ib[c]);
}

// ---------------- per-layer small gaussian precomputes ----------------
__global__ void pre_small_kernel(const float* __restrict__ sape, const float* __restrict__ tape,
                                 const float* __restrict__ trpe, const float* __restrict__ mu,
                                 const float* __restrict__ is, float* __restrict__ Gi,
                                 float* __restrict__ Egj, float* __restrict__ Egti,
                                 float* __restrict__ Ek) {
  int tid = blockIdx.x * 256 + threadIdx.x;
  if (tid < 512) {
    Gi[tid] = gaussd(sape + tid * 8, mu + 0, is + 0);
  } else if (tid < 1024) {
    int j = tid - 512;
    Egj[j] = expf(gaussd(sape + j * 8, mu + 8, is + 8));
  } else if (tid < 1120) {
    int bt = tid - 1024;
    Egti[bt] = expf(gaussd(tape + bt * 8, mu + 16, is + 16));
  } else if (tid < 1408) {
    int q = tid - 1120;
    int bt = q / 3, k = q % 3;
    int b = bt / TT, t = bt % TT;
    int ts = t + k - BETA;
    const float* e = (ts >= 0) ? (tape + (b * TT + ts) * 8) : nullptr;
    float gtj = gaussd(e, mu + 24, is + 24);
    float gtr = gaussd(trpe + k * 8, mu + 40, is + 40);
    Ek[bt * 3 + k] = expf(gtj + gtr);
  }
}

// Am[j, ki] = exp(ge[i,j] + Gi[i]) * range_mask[j, ki],  ki = k*512 + i
__global__ void pre_am_kernel(const float* __restrict__ srpe, const float* __restrict__ rmask,
                              const float* __restrict__ Gi, const float* __restrict__ mu4,
                              const float* __restrict__ is4, float* __restrict__ Am) {
  int idx = blockIdx.x * 256 + threadIdx.x;   // < 786432
  int j = idx / KTOT;
  int ki = idx - j * KTOT;
  int i = ki & 511;
  float g = gaussd(srpe + ((size_t)i * VV + j) * 8, mu4, is4);
  Am[idx] = expf(g + Gi[i]) * rmask[idx];
}

// SE[v,n] = sape[v,:] @ seproj_w ; TE[bt,n] = tape[bt,:] @ teproj_w
__global__ void pre_sete_kernel(const float* __restrict__ sape, const float* __restrict__ tape,
                                const float* __restrict__ sew, const float* __restrict__ tew,
                                float* __restrict__ SE, float* __restrict__ TE) {
  int idx = blockIdx.x * 256 + threadIdx.x;   // < 155648
  if (idx < VV * 256) {
    int v = idx >> 8, n = idx & 255;
    float s = 0.0f;
#pragma unroll
    for (int dd = 0; dd < DD; ++dd) s = fmaf(sape[v * 8 + dd], sew[dd * 256 + n], s);
    SE[idx] = s;
  } else {
    int q = idx - VV * 256;
    int bt = q >> 8, n = q & 255;
    float s = 0.0f;
#pragma unroll
    for (int dd = 0; dd < DD; ++dd) s = fmaf(tape[bt * 8 + dd], tew[dd * 256 + n], s);
    TE[q] = s;
  }
}

// pair-interleave xproj_w along K so zgemm B fragments are contiguous b64:
// xwi[(k>>1)*512 + n*2 + (k&1)] = xw[k*256 + n]
__global__ void pre_xwi_kernel(const float* __restrict__ xw, float* __restrict__ xwi) {
  int idx = blockIdx.x * 256 + threadIdx.x;   // < 32768
  int k = idx >> 8, n = idx & 255;
  xwi[(size_t)(k >> 1) * 512 + n * 2 + (k & 1)] = xw[idx];
}

// ---------------- main fused S*xwin aggregation GEMM ----------------
// agg[bt, j, c] = Egj[j]*Egti[bt] * sum_ki Am[j,ki]*Ek[bt,k]*xwin[bt,ki,c]
// grid (96, 4), block 256 (8 waves); wave -> one 16-row j-tile, all 128 c cols.
// B tile kept in LDS pair-interleaved: value(local row kk, col c) at Bsh[(kk>>1)*256 + c*2 + (kk&1)]
__global__ void agg_kernel(const float* __restrict__ h, const float* __restrict__ Am,
                           const float* __restrict__ Ek, const float* __restrict__ Egj,
                           const float* __restrict__ Egti, float* __restrict__ agg) {
  __shared__ float Bsh[32 * 256];
  int bt = blockIdx.x;
  int b = bt / TT, t = bt % TT;
  int wave = threadIdx.x >> 5;
  int lane = threadIdx.x & 31;
  int jtile = blockIdx.y * 8 + wave;            // 0..31
  int jA = jtile * 16 + (lane & 15);            // A-matrix row for this lane
  int khalf = (lane >> 4) * 2;                  // 0 or 2
  int kh2 = khalf >> 1;                         // 0 or 1
  int nlo = lane & 15;

  v8f acc[8];
#pragma unroll
  for (int i = 0; i < 8; ++i) acc[i] = zero8();

  for (int kc = 0; kc < KTOT / 64; ++kc) {
    int k = kc >> 3;                 // 64 | 512 -> k uniform within chunk
    int ts = t + k - BETA;
    if (ts < 0) continue;            // whole B tile zero: contributes nothing (uniform)
    float ek = Ek[bt * 3 + k];
    const float* hbase = h + (size_t)(b * TT + ts) * VV * CC;
    int ibase = (kc & 7) * 64;

    // stage 64 x 128 B tile, pair-interleaved, scaled by ek
#pragma unroll
    for (int it = 0; it < 8; ++it) {
      int slot = it * 256 + threadIdx.x;      // 0..2047
      int kkp = slot >> 6;                    // 0..31  (pair-row)
      int cp  = slot & 63;                    // 0..63  (col pair)
      int i0  = ibase + kkp * 2;
      int c   = cp * 2;
      v2f g0 = *(const v2f*)(hbase + (size_t)i0 * CC + c);
      v2f g1 = *(const v2f*)(hbase + (size_t)(i0 + 1) * CC + c);
      float4 w;
      w.x = g0.x * ek; w.y = g1.x * ek; w.z = g0.y * ek; w.w = g1.y * ek;
      *(float4*)&Bsh[kkp * 256 + cp * 4] = w;
    }

    // batch-preload A fragments for the whole chunk (one wait, pipelined)
    const float* AmRow = Am + (size_t)jA * KTOT + kc * 64 + khalf;
    v2f afr[16];
#pragma unroll
    for (int ks = 0; ks < 16; ++ks) afr[ks] = *(const v2f*)(AmRow + ks * 4);

    __syncthreads();

    // software-pipelined B fragments: prefetch kstep+1 while computing kstep
    v2f barr[8];
    {
      const float* brow = &Bsh[kh2 * 256 + nlo * 2];
#pragma unroll
      for (int ct = 0; ct < 8; ++ct) barr[ct] = *(const v2f*)(brow + ct * 32);
    }
#pragma unroll
    for (int ks = 0; ks < 16; ++ks) {
      v2f bnext[8];
      if (ks < 15) {
        const float* brow = &Bsh[((ks + 1) * 2 + kh2) * 256 + nlo * 2];
#pragma unroll
        for (int ct = 0; ct < 8; ++ct) bnext[ct] = *(const v2f*)(brow + ct * 32);
      }
#pragma unroll
      for (int ct = 0; ct < 8; ++ct) acc[ct] = wmma4(afr[ks], barr[ct], acc[ct]);
      if (ks < 15) {
#pragma unroll
        for (int ct = 0; ct < 8; ++ct) barr[ct] = bnext[ct];
      }
    }
    __syncthreads();
  }

  float egti = Egti[bt];
#pragma unroll
  for (int r = 0; r < 8; ++r) {
    int j = jtile * 16 + r + (lane >> 4) * 8;
    float scale = Egj[j] * egti;
    float* dst = agg + ((size_t)bt * VV + j) * CC;
#pragma unroll
    for (int ct = 0; ct < 8; ++ct) dst[ct * 16 + nlo] = acc[ct][r] * scale;
  }
}

// ---------------- z = agg @ xproj_w + xb + SE[v] + TE[bt] : M=49152,K=128,N=256 ----------------
__global__ void zgemm_kernel(const float* __restrict__ agg, const float* __restrict__ xwi,
                             const float* __restrict__ xb, const float* __restrict__ SE,
                             const float* __restrict__ TE, float* __restrict__ z) {
  int wave = threadIdx.x >> 5, lane = threadIdx.x & 31;
  int mtile = blockIdx.x * 8 + wave;            // < 3072
  int khalf = (lane >> 4) * 2, kh2 = khalf >> 1, nlo = lane & 15;
  const float* arow = agg + (size_t)(mtile * 16 + nlo) * CC + khalf;
  v8f acc[16];
#pragma unroll
  for (int i = 0; i < 16; ++i) acc[i] = zero8();

  v2f a_cur = *(const v2f*)(arow);
  v2f b_cur[16];
  {
    const float* b0 = xwi + (size_t)kh2 * 512 + nlo * 2;
#pragma unroll
    for (int nt = 0; nt < 16; ++nt) b_cur[nt] = *(const v2f*)(b0 + nt * 32);
  }
  for (int ks = 0; ks < 31; ++ks) {
    v2f a_nxt = *(const v2f*)(arow + (ks + 1) * 4);
    v2f b_nxt[16];
    const float* b0 = xwi + (size_t)((ks + 1) * 2 + kh2) * 512 + nlo * 2;
#pragma unroll
    for (int nt = 0; nt < 16; ++nt) b_nxt[nt] = *(const v2f*)(b0 + nt * 32);
#pragma unroll
    for (int nt = 0; nt < 16; ++nt) acc[nt] = wmma4(a_cur, b_cur[nt], acc[nt]);
    a_cur = a_nxt;
#pragma unroll
    for (int nt = 0; nt < 16; ++nt) b_cur[nt] = b_nxt[nt];
  }
#pragma unroll
  for (int nt = 0; nt < 16; ++nt) acc[nt] = wmma4(a_cur, b_cur[nt], acc[nt]);

  int mbase = mtile * 16 + (lane >> 4) * 8;
#pragma unroll
  for (int r = 0; r < 8; ++r) {
    int m = mbase + r;
    int v = m & 511, bt = m >> 9;
#pragma unroll
    for (int nt = 0; nt < 16; ++nt) {
      int n = nt * 16 + nlo;
      z[(size_t)m * 256 + n] = acc[nt][r] + xb[n] + SE[v * 256 + n] + TE[bt * 256 + n];
    }
  }
}

// ---------------- LayerNorm(256) + GLU -> h[b,t,v,0..127]; one wave per row ----------------
__global__ void ln_glu_kernel(const float* __restrict__ z, const float* __restrict__ lg,
                              const float* __restrict__ lb, float* __restrict__ hout) {
  int w = (blockIdx.x * 256 + threadIdx.x) >> 5;   // row m < 49152
  int lane = threadIdx.x & 31;
  const float* zr = z + (size_t)w * 256;
  float vals[8];
  float sum = 0.0f;
#pragma unroll
  for (int s = 0; s < 8; ++s) { vals[s] = zr[lane + 32 * s]; sum += vals[s]; }
#pragma unroll
  for (int off = 16; off > 0; off >>= 1) sum += __shfl_xor(sum, off, 32);
  float mean = sum * (1.0f / 256.0f);
  float var = 0.0f;
#pragma unroll
  for (int s = 0; s < 8; ++s) { float d = vals[s] - mean; var = fmaf(d, d, var); }
#pragma unroll
  for (int off = 16; off > 0; off >>= 1) var += __shfl_xor(var, off, 32);
  float rs = rsqrtf(var * (1.0f / 256.0f) + 1e-5f);
#pragma unroll
  for (int s = 0; s < 8; ++s) {
    int n = lane + 32 * s;
    vals[s] = fmaf((vals[s] - mean) * rs, lg[n], lb[n]);
  }
#pragma unroll
  for (int s = 0; s < 4; ++s) {
    int c = lane + 32 * s;
    hout[(size_t)w * CC + c] = vals[s] * sigmf(vals[s + 4]);
  }
}

// ---------------- GFS stage 1: s2[(b,v), o] = sum_{t,c} h[b,t,v,c]*fw[o,t,c] + fb[o] ----------------
// M=4096, K=1536, N=128
__global__ void gfs_s_kernel(const float* __restrict__ h, const float* __restrict__ fw,
                             const float* __restrict__ fb, float* __restrict__ s2) {
  int wave = threadIdx.x >> 5, lane = threadIdx.x & 31;
  int mtile = blockIdx.x * 8 + wave;            // < 256
  int khalf = (lane >> 4) * 2, nlo = lane & 15;
  int m = mtile * 16 + nlo;
  int b = m >> 9, v = m & 511;
  const float* ab = h + (size_t)b * (TT * VV * CC) + (size_t)v * CC;
  v8f acc[8];
#pragma unroll
  for (int i = 0; i < 8; ++i) acc[i] = zero8();

  v2f a_cur = *(const v2f*)(ab + khalf);
  v2f b_cur[8];
#pragma unroll
  for (int nt = 0; nt < 8; ++nt)
    b_cur[nt] = *(const v2f*)(fw + (size_t)(nt * 16 + nlo) * KTOT + khalf);

  for (int ks = 0; ks < 383; ++ks) {
    int tc = (ks + 1) * 4 + khalf;              // even -> pair stays within one t
    int t = tc >> 7, c = tc & 127;
    v2f a_nxt = *(const v2f*)(ab + (size_t)t * (VV * CC) + c);
    v2f b_nxt[8];
#pragma unroll
    for (int nt = 0; nt < 8; ++nt)
      b_nxt[nt] = *(const v2f*)(fw + (size_t)(nt * 16 + nlo) * KTOT + tc);
#pragma unroll
    for (int nt = 0; nt < 8; ++nt) acc[nt] = wmma4(a_cur, b_cur[nt], acc[nt]);
    a_cur = a_nxt;
#pragma unroll
    for (int nt = 0; nt < 8; ++nt) b_cur[nt] = b_nxt[nt];
  }
#pragma unroll
  for (int nt = 0; nt < 8; ++nt) acc[nt] = wmma4(a_cur, b_cur[nt], acc[nt]);

  int mbase = mtile * 16 + (lane >> 4) * 8;
#pragma unroll
  for (int r = 0; r < 8; ++r) {
#pragma unroll
    for (int nt = 0; nt < 8; ++nt) {
      int o = nt * 16 + nlo;
      s2[(size_t)(mbase + r) * CC + o] = acc[nt][r] + fb[o];
    }
  }
}

// ---------------- GFS stage 2 + GLU: hcT[(b*512+v), skip*128+p] ----------------
__global__ void gfs_g_kernel(const float* __restrict__ s2, const float* __restrict__ gw,
                             const float* __restrict__ gb, float* __restrict__ hcT, int skipIdx) {
  int wave = threadIdx.x >> 5, lane = threadIdx.x & 31;
  int mtile = blockIdx.x * 8 + wave;            // < 256
  int khalf = (lane >> 4) * 2, nlo = lane & 15;
  const float* arow = s2 + (size_t)(mtile * 16 + nlo) * CC + khalf;
  v8f acc[16];
#pragma unroll
  for (int i = 0; i < 16; ++i) acc[i] = zero8();

  v2f a_cur = *(const v2f*)(arow);
  v2f b_cur[16];
#pragma unroll
  for (int nt = 0; nt < 16; ++nt)
    b_cur[nt] = *(const v2f*)(gw + (size_t)(nt * 16 + nlo) * CC + khalf);

  for (int ks = 0; ks < 31; ++ks) {
    int o0 = (ks + 1) * 4 + khalf;
    v2f a_nxt = *(const v2f*)(arow + (ks + 1) * 4);
    v2f b_nxt[16];
#pragma unroll
    for (int nt = 0; nt < 16; ++nt)
      b_nxt[nt] = *(const v2f*)(gw + (size_t)(nt * 16 + nlo) * CC + o0);
#pragma unroll
    for (int nt = 0; nt < 16; ++nt) acc[nt] = wmma4(a_cur, b_cur[nt], acc[nt]);
    a_cur = a_nxt;
#pragma unroll
    for (int nt = 0; nt < 16; ++nt) b_cur[nt] = b_nxt[nt];
  }
#pragma unroll
  for (int nt = 0; nt < 16; ++nt) acc[nt] = wmma4(a_cur, b_cur[nt], acc[nt]);

  int mbase = mtile * 16 + (lane >> 4) * 8;
#pragma unroll
  for (int r = 0; r < 8; ++r) {
    int m = mbase + r;
#pragma unroll
    for (int pt = 0; pt < 8; ++pt) {
      int p = pt * 16 + nlo;
      float lhs = acc[pt][r] + gb[p];
      float rhs = acc[pt + 8][r] + gb[p + 128];
      hcT[(size_t)m * 512 + skipIdx * CC + p] = lhs * sigmf(rhs);
    }
  }
}

// ---------------- final big GLU GEMM: g2[(b,v), p] = sum_o glu_w[p,o]*hcT[(b,v),o] + glu_b[p] ----------------
// M=4096, K=512, N=1024 (grid.y = 4 groups of 256 cols)
__global__ void glugemm_kernel(const float* __restrict__ hcT, const float* __restrict__ gw,
                               const float* __restrict__ gb, float* __restrict__ g2) {
  int wave = threadIdx.x >> 5, lane = threadIdx.x & 31;
  int mtile = blockIdx.x * 8 + wave;            // < 256
  int ng = blockIdx.y;                          // 0..3
  int khalf = (lane >> 4) * 2, nlo = lane & 15;
  const float* arow = hcT + (size_t)(mtile * 16 + nlo) * 512 + khalf;
  const float* gwb = gw + (size_t)(ng * 256 + nlo) * 512 + khalf;
  v8f acc[16];
#pragma unroll
  for (int i = 0; i < 16; ++i) acc[i] = zero8();

  v2f a_cur = *(const v2f*)(arow);
  v2f b_cur[16];
#pragma unroll
  for (int nt = 0; nt < 16; ++nt) b_cur[nt] = *(const v2f*)(gwb + (size_t)nt * 16 * 512);

  for (int ks = 0; ks < 127; ++ks) {
    v2f a_nxt = *(const v2f*)(arow + (ks + 1) * 4);
    v2f b_nxt[16];
#pragma unroll
    for (int nt = 0; nt < 16; ++nt)
      b_nxt[nt] = *(const v2f*)(gwb + (size_t)nt * 16 * 512 + (ks + 1) * 4);
#pragma unroll
    for (int nt = 0; nt < 16; ++nt) acc[nt] = wmma4(a_cur, b_cur[nt], acc[nt]);
    a_cur = a_nxt;
#pragma unroll
    for (int nt = 0; nt < 16; ++nt) b_cur[nt] = b_nxt[nt];
  }
#pragma unroll
  for (int nt = 0; nt < 16; ++nt) acc[nt] = wmma4(a_cur, b_cur[nt], acc[nt]);

  int mbase = mtile * 16 + (lane >> 4) * 8;
#pragma unroll
  for (int r = 0; r < 8; ++r) {
    int m2 = mbase + r;
#pragma unroll
    for (int nt = 0; nt < 16; ++nt) {
      int p = ng * 256 + nt * 16 + nlo;
      g2[(size_t)m2 * 1024 + p] = acc[nt][r] + gb[p];
    }
  }
}

// ---------------- final GLU + out projection: out[b,p,v] ----------------
__global__ void out_kernel(const float* __restrict__ g2, const float* __restrict__ ow,
                           const float* __restrict__ ob, float* __restrict__ out) {
  int w = (blockIdx.x * 256 + threadIdx.x) >> 5;  // wave id < 49152
  int lane = threadIdx.x & 31;
  int m = w / PP, p = w % PP;
  const float* row = g2 + (size_t)m * 1024;
  const float* wr = ow + p * 512;
  float acc = 0.0f;
  for (int q = lane; q < 512; q += 32) {
    float l = row[q], r = row[q + 512];
    acc = fmaf(wr[q], l * sigmf(r), acc);
  }
#pragma unroll
  for (int off = 16; off > 0; off >>= 1) acc += __shfl_xor(acc, off, 32);
  if (lane == 0) {
    int b = m >> 9, v = m & 511;
    out[(size_t)b * (PP * VV) + p * VV + v] = acc + ob[p];
  }
}

extern "C" void kernel_launch(void* const* d_in, const int* in_sizes, int n_in,
                              void* d_out, int out_size, void* d_ws, size_t ws_size,
                              hipStream_t stream) {
  (void)in_sizes; (void)n_in; (void)out_size; (void)ws_size;
  const float* x        = (const float*)d_in[0];
  const float* sape     = (const float*)d_in[1];
  const float* tape     = (const float*)d_in[2];
  const float* srpe     = (const float*)d_in[3];
  const float* trpe     = (const float*)d_in[4];
  const float* rmask    = (const float*)d_in[7];
  const float* input_w  = (const float*)d_in[8];
  const float* input_b  = (const float*)d_in[9];
  const float* fs0_fc_w = (const float*)d_in[10];
  const float* fs0_fc_b = (const float*)d_in[11];
  const float* fs0_gw   = (const float*)d_in[12];
  const float* fs0_gb   = (const float*)d_in[13];
  const float* mus      = (const float*)d_in[14];
  const float* invs     = (const float*)d_in[15];
  const float* xproj_w  = (const float*)d_in[16];
  const float* xproj_b  = (const float*)d_in[17];
  const float* seproj_w = (const float*)d_in[18];
  const float* teproj_w = (const float*)d_in[19];
  const float* ln_g     = (const float*)d_in[20];
  const float* ln_b     = (const float*)d_in[21];
  const float* fs_fc_w  = (const float*)d_in[22];
  const float* fs_fc_b  = (const float*)d_in[23];
  const float* fs_glu_w = (const float*)d_in[24];
  const float* fs_glu_b = (const float*)d_in[25];
  const float* glu_w    = (const float*)d_in[26];
  const float* glu_b    = (const float*)d_in[27];
  const float* out_w    = (const float*)d_in[28];
  const float* out_b    = (const float*)d_in[29];

  float* ws = (float*)d_ws;
  const size_t HSZ = (size_t)BB * TT * VV * CC;          // 6,291,456
  float* h0   = ws;
  float* h1   = h0 + HSZ;
  float* aggb = h1 + HSZ;                                // aliased by s2
  float* zbuf = aggb + HSZ;                              // 12,582,912; aliased by g2
  float* Am   = zbuf + (size_t)NBT * VV * 256;
  float* Gi   = Am + (size_t)VV * KTOT;
  float* Egj  = Gi + 512;
  float* Egti = Egj + 512;
  float* Ek   = Egti + 96;
  float* SE   = Ek + 288 + 64;                           // pad
  float* TE   = SE + (size_t)VV * 256;
  float* hcT  = TE + (size_t)NBT * 256;                  // [4096][512]
  float* xwi  = hcT + (size_t)4096 * 512;                // 32768
  // aliases (lifetimes don't overlap):
  float* s2 = aggb;   // GFS intermediate (524,288) reuses agg buffer
  float* g2 = zbuf;   // final GLU GEMM output (4,194,304) reuses z buffer

  // 1) input layer
  input_kernel<<<24576, 256, 0, stream>>>(x, input_w, input_b, h0);

  // 2) skip 0 (GFS on h0)
  gfs_s_kernel<<<32, 256, 0, stream>>>(h0, fs0_fc_w, fs0_fc_b, s2);
  gfs_g_kernel<<<32, 256, 0, stream>>>(s2, fs0_gw, fs0_gb, hcT, 0);

  float* hcur = h0;
  float* hnxt = h1;
  for (int l = 0; l < LL; ++l) {
    const float* mu = mus + l * 48;
    const float* is = invs + l * 48;
    pre_small_kernel<<<6, 256, 0, stream>>>(sape, tape, trpe, mu, is, Gi, Egj, Egti, Ek);
    pre_am_kernel<<<3072, 256, 0, stream>>>(srpe, rmask, Gi, mu + 32, is + 32, Am);
    pre_sete_kernel<<<608, 256, 0, stream>>>(sape, tape, seproj_w + l * 2048,
                                             teproj_w + l * 2048, SE, TE);
    pre_xwi_kernel<<<128, 256, 0, stream>>>(xproj_w + l * 32768, xwi);
    agg_kernel<<<dim3(NBT, 4), 256, 0, stream>>>(hcur, Am, Ek, Egj, Egti, aggb);
    zgemm_kernel<<<384, 256, 0, stream>>>(aggb, xwi, xproj_b + l * 256, SE, TE, zbuf);
    ln_glu_kernel<<<6144, 256, 0, stream>>>(zbuf, ln_g + l * 256, ln_b + l * 256, hnxt);
    gfs_s_kernel<<<32, 256, 0, stream>>>(hnxt, fs_fc_w + l * 196608, fs_fc_b + l * 128, s2);
    gfs_g_kernel<<<32, 256, 0, stream>>>(s2, fs_glu_w + l * 32768, fs_glu_b + l * 256,
                                         hcT, l + 1);
    float* tmp = hcur; hcur = hnxt; hnxt = tmp;
  }

  // 3) final GLU GEMM + output projection
  glugemm_kernel<<<dim3(32, 4), 256, 0, stream>>>(hcT, glu_w, glu_b, g2);
  out_kernel<<<6144, 256, 0, stream>>>(g2, out_w, out_b, (float*)d_out);
}